// IDWT2D_63969242906645
// MI455X (gfx1250) — compile-verified
//
#include <hip/hip_runtime.h>

// 3-level inverse 2D Haar-family DWT, fully fused.
// input_node: (64, 256, 256, 10) NHWC f32 ; lp, hp: 2-tap f32 ; out: (64, 512, 512) f32
//
// One-level synthesis (derived from the reference's upsample + cyclic conv):
//   out[2r+p, 2c+q] = lp[q]*(lp[p]*LL + hp[p]*TR) + hp[q]*(lp[p]*BL + hp[p]*BR)
// No boundary wrap survives the algebra -> pure local butterfly, 3 levels fused.

typedef __attribute__((ext_vector_type(2))) float v2f;
typedef __attribute__((ext_vector_type(4))) float v4f;
typedef __attribute__((ext_vector_type(8))) float v8f;

#if defined(__gfx1250__) && __has_builtin(__builtin_amdgcn_wmma_f32_16x16x4_f32)
#define HAVE_WMMA_F32_16X16X4 1
#else
#define HAVE_WMMA_F32_16X16X4 0
#endif

// ---------------------------------------------------------------------------
// Launched path: one thread computes a 4x4 output tile (fixed level-1 LL
// coordinate (r,c) in [0,128)^2).  19 scalar-equivalent loads -> 16 outputs;
// there is no cross-thread input reuse at the two fine levels, so direct
// loads (no LDS) are optimal.  Memory-bound: ~232 MB total traffic (the 40 B
// channel-interleaved pixels mean every input cache line is touched once),
// ~10 us at 23.3 TB/s.
// ---------------------------------------------------------------------------
__global__ __launch_bounds__(256) void idwt3_fused(
    const float* __restrict__ in, const float* __restrict__ lpw,
    const float* __restrict__ hpw, float* __restrict__ out) {
  const int tid = blockIdx.x * 256 + threadIdx.x;
  const int b = tid >> 14;          // 64 batches
  const int r = (tid >> 7) & 127;   // level-1 LL row
  const int c = tid & 127;          // level-1 LL col

  const float lp0 = lpw[0], lp1 = lpw[1];
  const float hp0 = hpw[0], hp1 = hpw[1];

  const float* __restrict__ pb = in + (size_t)b * (256u * 256u * 10u);

  // ---- level 2 (coarsest): A2 at (r,c); inputs ch0..3 @ (96+r/2, 96+c/2) ----
  float A2;
  {
    const int i3 = r >> 1, j3 = c >> 1;
    const float* p2 = pb + ((size_t)(96 + i3) * 256 + (96 + j3)) * 10;
    const v2f x01 = *(const v2f*)(p2);       // 40k bytes -> 8B aligned
    const v2f x23 = *(const v2f*)(p2 + 2);
    const float ra = (r & 1) ? lp1 : lp0;
    const float rd = (r & 1) ? hp1 : hp0;
    const float ca = (c & 1) ? lp1 : lp0;
    const float cd = (c & 1) ? hp1 : hp0;
    A2 = ca * (ra * x01.x + rd * x01.y) + cd * (ra * x23.x + rd * x23.y);
  }

  // ---- level 1: A1[p][q]; inputs ch4..6 @ (64+r, 64+c) ----
  float A1[2][2];
  {
    const float* p1 = pb + ((size_t)(64 + r) * 256 + (64 + c)) * 10;
    const v2f y45 = *(const v2f*)(p1 + 4);   // 40k+16 bytes -> 8B aligned
    const float y6 = p1[6];
    const float t0 = lp0 * A2 + hp0 * y45.x;      // row combine (parity 0)
    const float t1 = lp1 * A2 + hp1 * y45.x;      // row combine (parity 1)
    const float u0 = lp0 * y45.y + hp0 * y6;
    const float u1 = lp1 * y45.y + hp1 * y6;
    A1[0][0] = lp0 * t0 + hp0 * u0;
    A1[0][1] = lp1 * t0 + hp1 * u0;
    A1[1][0] = lp0 * t1 + hp0 * u1;
    A1[1][1] = lp1 * t1 + hp1 * u1;
  }

  // ---- level 0: 4x4 outputs; inputs ch7..9 @ (2r+p, 2c+q) ----
#pragma unroll
  for (int p = 0; p < 2; ++p) {
    const int u = 2 * r + p;
    const float* pz = pb + ((size_t)u * 256 + 2 * c) * 10;
    const float z7a = pz[7];
    const v2f z89a = *(const v2f*)(pz + 8);   // 40k+32 -> 8B aligned
    const float z7b = pz[17];
    const v2f z89b = *(const v2f*)(pz + 18);  // 40k+72 -> 8B aligned
#pragma unroll
    for (int a = 0; a < 2; ++a) {
      const float la = a ? lp1 : lp0;
      const float ha = a ? hp1 : hp0;
      const float m0 = la * A1[p][0] + ha * z7a;
      const float n0 = la * z89a.x + ha * z89a.y;
      const float m1 = la * A1[p][1] + ha * z7b;
      const float n1 = la * z89b.x + ha * z89b.y;
      v4f row;
      row.x = lp0 * m0 + hp0 * n0;
      row.y = lp1 * m0 + hp1 * n0;
      row.z = lp0 * m1 + hp0 * n1;
      row.w = lp1 * m1 + hp1 * n1;
      // output row 4r+2p+a, cols 4c..4c+3 (16B aligned); output is write-once
      v4f* dst = (v4f*)(out + ((size_t)(b * 512 + 4 * r + 2 * p + a) * 512 + 4 * c));
      __builtin_nontemporal_store(row, dst);
    }
  }
}

// ---------------------------------------------------------------------------
// Alternate WMMA pipeline (compiled alternate path): the per-level combine is
// a linear map, so a 16x16 tile of the finest level is Out = K * Zloc * K^T
// with
//   K[2i+p][i] = lp[p], K[2i+p][8+i] = hp[p]   (two nonzeros per row)
// and Zloc = [[A1(8x8), ch8(8x8)], [ch7(8x8), ch9(8x8)]].
// K is generated in registers (A fragment layout per ISA 7.12.2); the middle
// transpose goes through LDS (D layout -> A layout).  8x
// v_wmma_f32_16x16x4_f32 per tile.  Branch-free gather: K<8 is compile-time
// per unrolled chunk (t<2), and the detail-row channel is 7 + 2*(nl>>3).
// ---------------------------------------------------------------------------
#if HAVE_WMMA_F32_16X16X4
__device__ __forceinline__ float kl_entry(int m, int k, float lp0, float lp1,
                                          float hp0, float hp1) {
  const int i = m >> 1;
  float v = 0.0f;
  if (k == i) v = (m & 1) ? lp1 : lp0;
  if (k == 8 + i) v = (m & 1) ? hp1 : hp0;
  return v;
}

__global__ __launch_bounds__(32) void idwt_level0_wmma(
    const float* __restrict__ in, const float* __restrict__ A1g,
    const float* __restrict__ lpw, const float* __restrict__ hpw,
    float* __restrict__ out) {
  __shared__ float sm[16][17];
  const int l = threadIdx.x;   // one wave32 per block; EXEC all ones
  const int nl = l & 15;
  const int hi = l >> 4;
  const int tile = blockIdx.x;          // b*1024 + TI*32 + TJ
  const int b = tile >> 10;
  const int TI = (tile >> 5) & 31;
  const int TJ = tile & 31;
  const float lp0 = lpw[0], lp1 = lpw[1], hp0 = hpw[0], hp1 = hpw[1];
  const float* pb = in + (size_t)b * (256u * 256u * 10u);
  const float* pa = A1g + (size_t)b * (256u * 256u);

  const int col = 8 * TJ + (nl & 7);
  const int chd = 7 + 2 * (nl >> 3);    // detail rows: ch7 (nl<8) / ch9 (nl>=8)

  // T1 = K * Zloc   (K = 16, as 4 chained 16x16x4 WMMAs)
  v8f acc = {};
#pragma unroll
  for (int t = 0; t < 4; ++t) {
    const int k0 = 4 * t + 2 * hi;
    v2f av, bv;
    // A fragment (ISA layout): lane = M, VGPR0/1 = K0/K0+1 (+2 for hi half)
    av.x = kl_entry(nl, k0, lp0, lp1, hp0, hp1);
    av.y = kl_entry(nl, k0 + 1, lp0, lp1, hp0, hp1);
    // B fragment (transposed-symmetry layout): element (K = k0+g, N = nl)
    float zv[2];
#pragma unroll
    for (int g = 0; g < 2; ++g) {
      const int row = 8 * TI + ((4 * (t & 1)) + g + 2 * hi);  // (k0+g) & 7
      if (t < 2) {
        // approx rows: LL = A1 (nl<8) else BL = ch8; one divergent ptr select
        const float* src = (nl < 8)
                               ? (pa + (size_t)row * 256 + col)
                               : (pb + ((size_t)row * 256 + col) * 10 + 8);
        zv[g] = *src;
      } else {
        // detail rows: TR = ch7 (nl<8) else BR = ch9; pure address arithmetic
        zv[g] = pb[((size_t)row * 256 + col) * 10 + chd];
      }
    }
    bv.x = zv[0];
    bv.y = zv[1];
    acc = __builtin_amdgcn_wmma_f32_16x16x4_f32(false, av, false, bv, (short)0,
                                                acc, false, false);
  }

  // D layout -> LDS -> A layout (transpose of T1 for the second GEMM)
#pragma unroll
  for (int rg = 0; rg < 8; ++rg) sm[rg + 8 * hi][nl] = acc[rg];
  __syncthreads();

  // Out = T1 * K^T
  v8f acc2 = {};
#pragma unroll
  for (int t = 0; t < 4; ++t) {
    const int k0 = 4 * t + 2 * hi;
    v2f av, bv;
    av.x = sm[nl][k0];
    av.y = sm[nl][k0 + 1];
    bv.x = kl_entry(nl, k0, lp0, lp1, hp0, hp1);   // (K^T)(K,N) = K[N][K]
    bv.y = kl_entry(nl, k0 + 1, lp0, lp1, hp0, hp1);
    acc2 = __builtin_amdgcn_wmma_f32_16x16x4_f32(false, av, false, bv, (short)0,
                                                 acc2, false, false);
  }

  // D layout store: VGPR rg, lanes lo/hi -> row rg / rg+8, col = lane
#pragma unroll
  for (int rg = 0; rg < 8; ++rg) {
    out[((size_t)(b * 512) + 16 * TI + rg + 8 * hi) * 512 + 16 * TJ + nl] =
        acc2[rg];
  }
}
#endif  // HAVE_WMMA_F32_16X16X4

extern "C" void kernel_launch(void* const* d_in, const int* in_sizes, int n_in,
                              void* d_out, int out_size, void* d_ws, size_t ws_size,
                              hipStream_t stream) {
  (void)in_sizes; (void)n_in; (void)d_ws; (void)ws_size; (void)out_size;
  const float* in = (const float*)d_in[0];   // (64,256,256,10) f32
  const float* lp = (const float*)d_in[1];   // 2-tap lowpass
  const float* hp = (const float*)d_in[2];   // 2-tap highpass
  float* out = (float*)d_out;                // (64,512,512) f32

  // 64 * 128 * 128 threads, one 4x4 output tile each.
  idwt3_fused<<<dim3(4096), dim3(256), 0, stream>>>(in, lp, hp, out);
}